// DistanceWeightedMiner_80951543595527
// MI455X (gfx1250) — compile-verified
//
#include <hip/hip_runtime.h>
#include <math.h>

// ---------------- problem constants (match reference) ----------------
#define NROW 8192
#define DIM  128
#define NUM_CLASSES 256
#define MAX_K 64
#define MIN_DIST 0.1f
#define MAX_DIST 1.42f
// log_weights(d) = -( (D-2)*ln d + (D-3)/2 * (1 - d^2/4) ), D=128
#define LW_A 126.0f   // D-2
#define LW_B 62.5f    // (D-3)/2

typedef __attribute__((ext_vector_type(2))) float v2f;
typedef __attribute__((ext_vector_type(8))) float v8f;
typedef __attribute__((ext_vector_type(4))) unsigned u32x4;
typedef __attribute__((ext_vector_type(8))) unsigned u32x8;

static __device__ __forceinline__ v8f wmma_f32_k4(v2f a, v2f b, v8f c) {
  // V_WMMA_F32_16X16X4_F32 : D = A(16x4) x B(4x16) + C(16x16), all f32
  return __builtin_amdgcn_wmma_f32_16x16x4_f32(
      /*neg_a=*/false, a, /*neg_b=*/false, b,
      /*c_mod=*/(short)0, c, /*reuse_a=*/false, /*reuse_b=*/false);
}

static __device__ __forceinline__ float logw_of_dist(float d) {
  return -(LW_A * __logf(d) + LW_B * (1.0f - 0.25f * d * d));
}

static __device__ __forceinline__ unsigned rng32(unsigned x) {
  x += 0x9e3779b9u;
  x ^= x >> 16; x *= 0x21f0aaadu;
  x ^= x >> 15; x *= 0x735a2d97u;
  x ^= x >> 15;
  return x;
}

// Build TDM D# group0/group1 (CDNA5 ISA) for a 2D
// 64-row x 128-col f32 tile: global (row-major, stride DIM) -> LDS (packed).
static __device__ __forceinline__ u32x4 tdm_group0(unsigned lds_byte_off,
                                                   const float* tile_base) {
  unsigned long long ga = (unsigned long long)tile_base;  // 57-bit byte addr
  u32x4 g0;
  g0.x = 1u;                                   // count=1 (valid user D#)
  g0.y = lds_byte_off;                         // lds_addr [63:32]
  g0.z = (unsigned)(ga & 0xFFFFFFFFu);         // global_addr[31:0]
  g0.w = (unsigned)((ga >> 32) & 0x1FFFFFFu)   // global_addr[56:32]
         | (2u << 30);                         // type=2 ("image") [127:126]
  return g0;
}

static __device__ __forceinline__ u32x8 tdm_group1() {
  u32x8 g1;
  g1.s0 = (2u << 16);                          // wg_mask=0, data_size=2 (4B)
  g1.s1 = ((unsigned)DIM & 0xFFFFu) << 16;     // tensor_dim0[15:0] @ bits 63:48
  g1.s2 = (((unsigned)DIM >> 16) & 0xFFFFu)    // tensor_dim0[31:16]
          | (((unsigned)NROW & 0xFFFFu) << 16);// tensor_dim1[15:0]
  g1.s3 = (((unsigned)NROW >> 16) & 0xFFFFu)   // tensor_dim1[31:16]
          | ((unsigned)DIM << 16);             // tile_dim0 = 128 elements
  g1.s4 = 64u;                                 // tile_dim1 = 64 rows, tile_dim2=0
  g1.s5 = (unsigned)DIM;                       // tensor_dim0_stride[31:0] = 128
  g1.s6 = 0u;                                  // stride0[47:32] | dim1_stride lo
  g1.s7 = 0u;                                  // dim1_stride hi (unused, 2D)
  return g1;
}

// ---------------- 1: init scratch ----------------
__global__ void init_kernel(unsigned* minD, float* rowsum, int* counts) {
  int i = blockIdx.x * blockDim.x + threadIdx.x;
  if (i < NROW) {
    minD[i] = 0x7f7fffffu;  // +FLT_MAX bits (positive floats order as uints)
    rowsum[i] = 0.0f;
  }
  if (i < NUM_CLASSES) counts[i] = 0;
}

// ---------------- 2: L2-normalize rows (one wave32 per row) ----------------
__global__ __launch_bounds__(256) void normalize_kernel(const float* __restrict__ x,
                                                        float* __restrict__ xn) {
  int lane = threadIdx.x & 31;
  int wave = threadIdx.x >> 5;
  int row = blockIdx.x * 8 + wave;
  const float4 v = *(const float4*)&x[(size_t)row * DIM + lane * 4];
  float ss = v.x * v.x + v.y * v.y + v.z * v.z + v.w * v.w;
  #pragma unroll
  for (int off = 16; off >= 1; off >>= 1) ss += __shfl_xor(ss, off, 32);
  float inv = 1.0f / sqrtf(ss);
  float4 o = {v.x * inv, v.y * inv, v.z * inv, v.w * inv};
  *(float4*)&xn[(size_t)row * DIM + lane * 4] = o;
}

// ---------------- 3: label bincount ----------------
__global__ void bincount_kernel(const int* __restrict__ labels, int* counts) {
  int i = blockIdx.x * blockDim.x + threadIdx.x;
  if (i < NROW) atomicAdd(&counts[labels[i]], 1);
}

// ---------------- 4/6: WMMA gram-matrix passes (TDM-staged tiles) ----------
// PASS 0: per-row min distance (for the monotone logwmax trick)
// PASS 1: emit unnormalized w, pos_mask, accumulate row sums
template <int PASS>
__global__ __launch_bounds__(256) void gram_kernel(
    const float* __restrict__ xn, const int* __restrict__ labels,
    const float* __restrict__ logwmax, float* __restrict__ w_out,
    float* __restrict__ pos_out, unsigned* __restrict__ minD,
    float* __restrict__ rowsum) {
  __shared__ float lds[2 * 64 * 128];  // [0,32KB): A tile, [32KB,64KB): B tile
  float* lA = lds;
  float* lB = lds + 64 * 128;
  const int I = blockIdx.y * 64;
  const int J = blockIdx.x * 64;
  const int t = threadIdx.x;

  // Wave 0 issues two Tensor Data Mover descriptors: 64x128 f32 tiles of xn
  // DMA'd straight into LDS (TENSORcnt-tracked), then drains TENSORcnt.
  if (t < 32) {
    u32x4 g0a = tdm_group0(0u, xn + (size_t)I * DIM);
    u32x4 g0b = tdm_group0(64u * 128u * 4u, xn + (size_t)J * DIM);
    u32x8 g1 = tdm_group1();
    asm volatile("tensor_load_to_lds %0, %1" ::"s"(g0a), "s"(g1) : "memory");
    asm volatile("tensor_load_to_lds %0, %1" ::"s"(g0b), "s"(g1) : "memory");
    __builtin_amdgcn_s_wait_tensorcnt(0);
  }
  __syncthreads();

  const int lane = t & 31;
  const int wave = t >> 5;
  const int wr = wave & 3;   // which 16-row band
  const int wc = wave >> 2;  // which 32-col band
  const int l16 = lane & 15;
  const int hi = lane >> 4;

  const int m = (wr << 4) + l16;   // local A row for this lane
  const int n0 = (wc << 5) + l16;  // local B row (= gram column), subtile 0
  const int n1 = n0 + 16;          // subtile 1

  v8f acc0 = {}, acc1 = {};
  #pragma unroll 4
  for (int k = 0; k < DIM; k += 4) {
    const int k0 = k + (hi << 1);  // ISA f32 A layout: VGPR0=K{0|2}, VGPR1=K{1|3}
    v2f a  = {lA[m * 128 + k0],  lA[m * 128 + k0 + 1]};
    v2f b0 = {lB[n0 * 128 + k0], lB[n0 * 128 + k0 + 1]};
    v2f b1 = {lB[n1 * 128 + k0], lB[n1 * 128 + k0 + 1]};
    acc0 = wmma_f32_k4(a, b0, acc0);
    acc1 = wmma_f32_k4(a, b1, acc1);
  }

  // epilogue: C/D layout — VGPR r holds M=r (lanes 0-15) / M=r+8 (lanes 16-31)
  const int baseI = I + (wr << 4) + (hi << 3);
  #pragma unroll
  for (int sub = 0; sub < 2; ++sub) {
    const v8f acc = sub ? acc1 : acc0;
    const int gj = J + (wc << 5) + (sub ? 16 : 0) + l16;
    #pragma unroll
    for (int r = 0; r < 8; ++r) {
      const int gi = baseI + r;
      float g = acc[r];
      float dist = sqrtf(fmaxf(2.0f - 2.0f * g, 0.0f));
      if (gi == gj) dist = MAX_DIST;  // self-pair
      dist = fmaxf(dist, MIN_DIST);   // lower clip
      if (PASS == 0) {
        float dmin = dist;
        #pragma unroll
        for (int off = 8; off >= 1; off >>= 1)
          dmin = fminf(dmin, __shfl_xor(dmin, off, 16));
        if (l16 == 0) atomicMin(&minD[gi], __float_as_uint(dmin));
      } else {
        const int li = labels[gi], lj = labels[gj];
        float wv = 0.0f;
        if (li != lj && dist < MAX_DIST)
          wv = __expf(logw_of_dist(dist) - logwmax[gi]);
        const size_t o = (size_t)gi * NROW + gj;
        w_out[o] = wv;
        pos_out[o] = (li == lj && gi != gj) ? 1.0f : 0.0f;
        float s = wv;
        #pragma unroll
        for (int off = 8; off >= 1; off >>= 1) s += __shfl_xor(s, off, 16);
        if (l16 == 0) atomicAdd(&rowsum[gi], s);
      }
    }
  }
}

// ---------------- 5: per-row logwmax and k ----------------
__global__ void rowprep_kernel(const unsigned* __restrict__ minD,
                               const int* __restrict__ labels,
                               const int* __restrict__ counts,
                               float* __restrict__ logwmax,
                               int* __restrict__ kArr) {
  int i = blockIdx.x * blockDim.x + threadIdx.x;
  if (i >= NROW) return;
  float mind = __uint_as_float(minD[i]);  // logw monotone-decreasing on (0,2]
  logwmax[i] = logw_of_dist(mind);
  kArr[i] = counts[labels[i]] - 1;
}

// ---------------- 7: row-normalize w, zero neg_mask ----------------
__global__ void finalize_w_kernel(float* __restrict__ w_out,
                                  float* __restrict__ neg_out,
                                  const float* __restrict__ rowsum) {
  size_t idx = (size_t)blockIdx.x * blockDim.x + threadIdx.x;
  int row = (int)(idx >> 13);  // /NROW
  float inv = 1.0f / fmaxf(rowsum[row], 1e-6f);
  w_out[idx] *= inv;
  neg_out[idx] = 0.0f;
}

// ---------------- 8: inverse-CDF categorical sampling (one block per row) ---
__global__ __launch_bounds__(256) void sample_kernel(const float* __restrict__ w,
                                                     const int* __restrict__ kArr,
                                                     float* __restrict__ neg_out) {
  __shared__ float cdf[NROW];  // 32 KB — trivial for a 320 KB-LDS WGP
  __shared__ float psum[257];
  __shared__ float sTotal;
  const int row = blockIdx.x;
  const int t = threadIdx.x;
  const int base = t * 32;

  float csum = 0.0f;
  #pragma unroll 4
  for (int j = 0; j < 32; ++j) {
    float v = w[(size_t)row * NROW + base + j];
    cdf[base + j] = v;
    csum += v;
  }
  psum[t] = csum;
  __syncthreads();
  if (t == 0) {  // tiny serial exclusive scan of 256 chunk sums
    float run = 0.0f;
    for (int i = 0; i < 256; ++i) { float tmp = psum[i]; psum[i] = run; run += tmp; }
    sTotal = run;
  }
  __syncthreads();
  float run = psum[t];
  #pragma unroll 4
  for (int j = 0; j < 32; ++j) { run += cdf[base + j]; cdf[base + j] = run; }
  __syncthreads();

  const float total = sTotal;
  const int k = min(kArr[row], MAX_K);
  if (t < k && total > 0.0f) {
    float u = (rng32((unsigned)(row * MAX_K + t) ^ 0xC0FFEEu) >> 8) *
              (1.0f / 16777216.0f) * total;
    int lo = 0, hiB = NROW - 1;
    while (lo < hiB) {  // first idx with cdf[idx] > u
      int mid = (lo + hiB) >> 1;
      if (cdf[mid] <= u) lo = mid + 1; else hiB = mid;
    }
    neg_out[(size_t)row * NROW + lo] = 1.0f;
  }
}

// ---------------- launch ----------------
extern "C" void kernel_launch(void* const* d_in, const int* in_sizes, int n_in,
                              void* d_out, int out_size, void* d_ws, size_t ws_size,
                              hipStream_t stream) {
  const float* embeds = (const float*)d_in[0];
  const int* labels = (const int*)d_in[1];

  float* w_out = (float*)d_out;
  float* pos_out = w_out + (size_t)NROW * NROW;
  float* neg_out = pos_out + (size_t)NROW * NROW;

  // workspace layout (~4.3 MB)
  float* xn = (float*)d_ws;                       // N*D normalized embeds
  unsigned* minD = (unsigned*)(xn + NROW * DIM);  // N
  float* rowsum = (float*)(minD + NROW);          // N
  float* logwmax = rowsum + NROW;                 // N
  int* counts = (int*)(logwmax + NROW);           // NUM_CLASSES
  int* kArr = counts + NUM_CLASSES;               // N

  init_kernel<<<NROW / 256, 256, 0, stream>>>(minD, rowsum, counts);
  normalize_kernel<<<NROW / 8, 256, 0, stream>>>(embeds, xn);
  bincount_kernel<<<NROW / 256, 256, 0, stream>>>(labels, counts);

  dim3 gemm_grid(NROW / 64, NROW / 64);
  gram_kernel<0><<<gemm_grid, 256, 0, stream>>>(xn, labels, logwmax, w_out,
                                                pos_out, minD, rowsum);
  rowprep_kernel<<<NROW / 256, 256, 0, stream>>>(minD, labels, counts, logwmax, kArr);
  gram_kernel<1><<<gemm_grid, 256, 0, stream>>>(xn, labels, logwmax, w_out,
                                                pos_out, minD, rowsum);

  finalize_w_kernel<<<(unsigned)(((size_t)NROW * NROW) / 256), 256, 0, stream>>>(
      w_out, neg_out, rowsum);
  sample_kernel<<<NROW, 256, 0, stream>>>(w_out, kArr, neg_out);
}